// Sampler_54494545052319
// MI455X (gfx1250) — compile-verified
//
#include <hip/hip_runtime.h>
#include <hip/hip_bf16.h>
#include <stdint.h>

// ---------------------------------------------------------------------------
// Gumbel-max sampler, single streaming pass (memory-bound: 262 MB @ 23.3 TB/s).
//   greedy  = argmax_v logits[v]
//   sample  = argmax_v ( logits[v]/t - log(-log(1-u[v]) + 1e-10) )
//   out     = (t == 0) ? greedy : sample
// Softmax normalization (row max + row sum) is argmax-invariant -> dropped.
// Data movement: gfx1250 async global->LDS (ASYNCcnt) 4-stage pipeline,
// steady-state loop with a constant s_wait_asynccnt immediate.
// ---------------------------------------------------------------------------

#define BLOCK      640                    // 20 wave32 waves
#define WAVES      (BLOCK / 32)
#define TILE_ELEMS (BLOCK * 4)            // 2560 floats / tile (float4 per lane)
#define TILE_BYTES (TILE_ELEMS * 4)       // 10240 B
#define STAGES     4                      // tiles resident in LDS

// One async 16B/lane transfer: global saddr plus voff_bytes into LDS at byte address lds_byte_addr
// (GVS mode: 64-bit SGPR base + 32-bit per-lane VGPR byte offset.)
__device__ __forceinline__ void async_load16(unsigned lds_byte_addr,
                                             unsigned voff_bytes,
                                             const float* saddr) {
  asm volatile("global_load_async_to_lds_b128 %0, %1, %2"
               :
               : "v"(lds_byte_addr), "v"(voff_bytes), "s"(saddr)
               : "memory");
}

// In-thread running max: indices strictly increase, so '>' alone preserves
// first-occurrence semantics (cheapest possible: 1 cmp + 2 cndmask).
__device__ __forceinline__ void takeMax(float v, int i, float& bv, int& bi) {
  if (v > bv) { bv = v; bi = i; }
}

// Cross-lane combine: needs the explicit lower-index tie-break.
__device__ __forceinline__ void comb(float v, int i, float& bv, int& bi) {
  if (v > bv || (v == bv && i < bi)) { bv = v; bi = i; }
}

__device__ __forceinline__ void accum(float l, float u, int idx, float invt,
                                      float& gB, int& gI, float& sB, int& sI) {
  takeMax(l, idx, gB, gI);
  // score = l/t - log(-log(1-u) + eps); -log(1-u)+eps folded as eps - log(1-u)
  float lg    = __logf(1.0f - u);
  float score = l * invt - __logf(1e-10f - lg);
  takeMax(score, idx, sB, sI);
}

__global__ __launch_bounds__(BLOCK, 1)
void Sampler_54494545052319_kernel(const float* __restrict__ logits,
                                   const float* __restrict__ temps,
                                   const float* __restrict__ expu,
                                   int* __restrict__ out, int V) {
  __shared__ __align__(16) float ldsL[STAGES][TILE_ELEMS];
  __shared__ __align__(16) float ldsU[STAGES][TILE_ELEMS];
  __shared__ float wgv[2][WAVES];
  __shared__ int   wgi[2][WAVES];

  const int row = blockIdx.x;
  const int tid = threadIdx.x;

  const float t    = temps[row];
  const float invt = 1.0f / ((t == 0.0f) ? 1.0f : t);

  const float* rowL = logits + (size_t)row * (size_t)V;
  const float* rowU = expu   + (size_t)row * (size_t)V;

  // Flat LDS pointers: low 32 bits == byte offset within the WG's LDS alloc.
  const unsigned baseL  = (unsigned)(uintptr_t)&ldsL[0][0];
  const unsigned baseU  = (unsigned)(uintptr_t)&ldsU[0][0];
  const unsigned lane16 = (unsigned)tid * 16u;

  const int nTiles = V / TILE_ELEMS;

  float gBest = -INFINITY; int gIdx = 0x7fffffff;  // greedy
  float sBest = -INFINITY; int sIdx = 0x7fffffff;  // sampled

  // Prologue: prefetch up to STAGES-1 tiles ahead.
  const int pre = (nTiles < (STAGES - 1)) ? nTiles : (STAGES - 1);
  for (int p = 0; p < pre; ++p) {
    unsigned boff = (unsigned)(p & (STAGES - 1)) * TILE_BYTES + lane16;
    unsigned goff = (unsigned)p * TILE_BYTES + lane16;
    async_load16(baseL + boff, goff, rowL);
    async_load16(baseU + boff, goff, rowU);
  }

  // Steady state: always STAGES-1 tiles in flight -> constant wait immediate.
  const int steady = nTiles - (STAGES - 1);
  int tl = 0;
#pragma unroll 4
  for (; tl < steady; ++tl) {
    const int nx = tl + (STAGES - 1);
    unsigned boff = (unsigned)(nx & (STAGES - 1)) * TILE_BYTES + lane16;
    unsigned goff = (unsigned)nx * TILE_BYTES + lane16;
    async_load16(baseL + boff, goff, rowL);
    async_load16(baseU + boff, goff, rowU);

    // 3 tile-pairs may remain in flight; in-order completion => tile tl ready.
    asm volatile("s_wait_asynccnt 0x6" ::: "memory");

    const int buf = tl & (STAGES - 1);
    const float4 lv = *reinterpret_cast<const float4*>(&ldsL[buf][tid * 4]);
    const float4 uv = *reinterpret_cast<const float4*>(&ldsU[buf][tid * 4]);
    const int base = tl * TILE_ELEMS + tid * 4;

    accum(lv.x, uv.x, base + 0, invt, gBest, gIdx, sBest, sIdx);
    accum(lv.y, uv.y, base + 1, invt, gBest, gIdx, sBest, sIdx);
    accum(lv.z, uv.z, base + 2, invt, gBest, gIdx, sBest, sIdx);
    accum(lv.w, uv.w, base + 3, invt, gBest, gIdx, sBest, sIdx);
  }

  // Tail: drain the last (up to) STAGES-1 tiles; everything already issued.
  for (tl = (steady > 0 ? steady : 0); tl < nTiles; ++tl) {
    const int beyond = nTiles - 1 - tl;            // 2, 1, 0
    if (beyond >= 2)      asm volatile("s_wait_asynccnt 0x4" ::: "memory");
    else if (beyond == 1) asm volatile("s_wait_asynccnt 0x2" ::: "memory");
    else                  asm volatile("s_wait_asynccnt 0x0" ::: "memory");

    const int buf = tl & (STAGES - 1);
    const float4 lv = *reinterpret_cast<const float4*>(&ldsL[buf][tid * 4]);
    const float4 uv = *reinterpret_cast<const float4*>(&ldsU[buf][tid * 4]);
    const int base = tl * TILE_ELEMS + tid * 4;

    accum(lv.x, uv.x, base + 0, invt, gBest, gIdx, sBest, sIdx);
    accum(lv.y, uv.y, base + 1, invt, gBest, gIdx, sBest, sIdx);
    accum(lv.z, uv.z, base + 2, invt, gBest, gIdx, sBest, sIdx);
    accum(lv.w, uv.w, base + 3, invt, gBest, gIdx, sBest, sIdx);
  }

  // Remainder elements (dead for V = 128000, kept for generality).
  for (int i = nTiles * TILE_ELEMS + tid; i < V; i += BLOCK) {
    accum(rowL[i], rowU[i], i, invt, gBest, gIdx, sBest, sIdx);
  }

  // Wave32 tree reduction (val, idx) with tie-break, then cross-wave via LDS.
  for (int m = 16; m > 0; m >>= 1) {
    float gv = __shfl_xor(gBest, m, 32); int gi2 = __shfl_xor(gIdx, m, 32);
    comb(gv, gi2, gBest, gIdx);
    float sv = __shfl_xor(sBest, m, 32); int si2 = __shfl_xor(sIdx, m, 32);
    comb(sv, si2, sBest, sIdx);
  }

  const int lane = tid & 31, wid = tid >> 5;
  if (lane == 0) {
    wgv[0][wid] = gBest; wgi[0][wid] = gIdx;
    wgv[1][wid] = sBest; wgi[1][wid] = sIdx;
  }
  __syncthreads();
  if (tid == 0) {
    float bg = wgv[0][0]; int big = wgi[0][0];
    float bs = wgv[1][0]; int bis = wgi[1][0];
    for (int w = 1; w < WAVES; ++w) {
      comb(wgv[0][w], wgi[0][w], bg, big);
      comb(wgv[1][w], wgi[1][w], bs, bis);
    }
    out[row] = (t == 0.0f) ? big : bis;
  }
}

extern "C" void kernel_launch(void* const* d_in, const int* in_sizes, int n_in,
                              void* d_out, int out_size, void* d_ws, size_t ws_size,
                              hipStream_t stream) {
  const float* logits = (const float*)d_in[0];   // [B, V] f32
  const float* temps  = (const float*)d_in[1];   // [B]    f32
  const float* expu   = (const float*)d_in[2];   // [B, V] f32
  int*         out    = (int*)d_out;             // [B] token ids

  const int B = in_sizes[1];
  const int V = in_sizes[0] / B;

  Sampler_54494545052319_kernel<<<dim3(B), dim3(BLOCK), 0, stream>>>(
      logits, temps, expu, out, V);
}